// SAGE_27539330301989
// MI455X (gfx1250) — compile-verified
//
#include <hip/hip_runtime.h>
#include <hip/hip_bf16.h>

// GraphSAGE 3-layer forward for MI455X (gfx1250, wave32).
//
// Cost model:
//  - GEMMs: ~65 GFLOP total -> negligible; full-precision V_WMMA_F32_16X16X4_F32.
//  - Aggregation dominates: E*d random gathers + E*d f32 scatter atomics/layer.
//    Node features (51..102 MB) fit in the 192 MB L2, so gathers and
//    global_atomic_add_f32 (no-return, STOREcnt) resolve in L2.
//  - GEMM A-panel (16 rows x [agg*invdeg | x]) is staged once per block into
//    LDS (ds_store_b128), padded stride -> bank-conflict-free ds_load_b64
//    fragments shared by all 4 waves; inv_deg applied at staging time.

typedef __attribute__((ext_vector_type(2))) float v2f;
typedef __attribute__((ext_vector_type(8))) float v8f;

#define DHID 256
#define APAD 4                       // dword padding per LDS row (bank spread)
#define KMAX 512                     // max K = 2*din (layers 2/3)

// ---------------------------------------------------------------- utilities
__global__ void sage_zero_kernel(float* __restrict__ p, long long n) {
  long long i = (long long)blockIdx.x * blockDim.x + threadIdx.x;
  long long stride = (long long)gridDim.x * blockDim.x;
  for (; i < n; i += stride) p[i] = 0.0f;
}

__global__ void sage_degree_kernel(const int* __restrict__ dst,
                                   float* __restrict__ deg, int E) {
  int i = blockIdx.x * blockDim.x + threadIdx.x;
  if (i < E) atomicAdd(&deg[dst[i]], 1.0f);
}

__global__ void sage_invdeg_kernel(float* __restrict__ deg, int N) {
  int i = blockIdx.x * blockDim.x + threadIdx.x;
  if (i < N) deg[i] = 1.0f / fmaxf(deg[i], 1.0f);
}

// ------------------------------------------------- edge scatter (agg += x[src])
// One thread per (edge, 4-float chunk): coalesced 128b gather of the source
// row, scatter via 4x global_atomic_add_f32 resolving in L2.
__global__ void sage_scatter_kernel(const float* __restrict__ x,
                                    const int* __restrict__ src,
                                    const int* __restrict__ dst,
                                    float* __restrict__ agg,
                                    long long total, int d, int cshift) {
  long long idx = (long long)blockIdx.x * blockDim.x + threadIdx.x;
  if (idx >= total) return;
  int chunks_m1 = (1 << cshift) - 1;
  int e = (int)(idx >> cshift);
  int c = (int)(idx & chunks_m1) << 2;
  int s = src[e];
  int t = dst[e];
  const float4 v = *(const float4*)(x + (long long)s * d + c);
  float* o = agg + (long long)t * d + c;
  atomicAdd(o + 0, v.x);
  atomicAdd(o + 1, v.y);
  atomicAdd(o + 2, v.z);
  atomicAdd(o + 3, v.w);
}

// ------------------------------------------ fused SAGE GEMM via WMMA f32 16x16x4
// out[m,n] = sum_k agg[m,k]*invdeg[m]*Wl[k,n] + sum_k x[m,k]*Wr[k,n] + bias[n]
//
// Block = 128 threads (4 waves) -> 16 rows x 64 cols of output.
// Stage A-panel [agg*invdeg | x] (16 x K floats, K = 2*din) into LDS once,
// then each wave computes one 16x16 tile with K/4 chained WMMAs.
//
// Fragment layouts (ISA 7.12.2, 32-bit):
//   A 16x4 : lane l<16 row M=l holds (K0,K1); lanes 16-31 hold (K2,K3)
//   B 4x16 : v0 = row K0 (lanes 0-15, N=lane%16) / K2 (lanes 16-31); v1 = K1/K3
//   C/D    : VGPR r = row r (lanes 0-15) / row r+8 (lanes 16-31), col = lane%16
__global__ __launch_bounds__(128)
void sage_gemm_wmma_kernel(const float* __restrict__ agg,
                           const float* __restrict__ xin,
                           const float* __restrict__ invdeg,
                           const float* __restrict__ Wl,
                           const float* __restrict__ Wr,
                           const float* __restrict__ bias,
                           float* __restrict__ out,
                           int din, int dout, int kshift) {
  __shared__ float As[16 * (KMAX + APAD)];

  const int K = din << 1;            // 256 (layer 1) or 512 (layers 2/3)
  const int astride = K + APAD;
  const int rowBase = blockIdx.x * 16;

  // ---- stage A-panel into LDS: row-> [agg*invd (din) | x (din)] ----
  const int chunks = (16 * K) >> 2;            // float4 chunks
  const int cPerRow_m1 = (K >> 2) - 1;         // power of two
  for (int i = threadIdx.x; i < chunks; i += blockDim.x) {
    const int row   = i >> (kshift - 2);
    const int kcol  = (i & cPerRow_m1) << 2;
    const int gm    = rowBase + row;
    float4 v;
    if (kcol < din) {
      v = *(const float4*)(agg + (long long)gm * din + kcol);
      const float s = invdeg[gm];
      v.x *= s; v.y *= s; v.z *= s; v.w *= s;
    } else {
      v = *(const float4*)(xin + (long long)gm * din + (kcol - din));
    }
    *(float4*)(&As[row * astride + kcol]) = v;
  }
  __syncthreads();

  const int lane = threadIdx.x & 31;
  const int wave = threadIdx.x >> 5;
  const int half = lane >> 4;        // 0: K0/K1 lanes, 1: K2/K3 lanes
  const int l    = lane & 15;
  const int n    = (blockIdx.y * 4 + wave) * 16 + l;

  const float* __restrict__ arow = As + l * astride + 2 * half;

  v8f c = {};

  // [agg*invd] @ Wl   (k = 0 .. din)
  for (int k0 = 0; k0 < din; k0 += 4) {
    const float2 av = *(const float2*)(arow + k0);
    v2f a; a.x = av.x; a.y = av.y;
    const int kk = k0 + 2 * half;
    v2f b;
    b.x = Wl[(long long)kk * dout + n];
    b.y = Wl[(long long)(kk + 1) * dout + n];
    c = __builtin_amdgcn_wmma_f32_16x16x4_f32(false, a, false, b,
                                              (short)0, c, false, false);
  }
  // x @ Wr            (k = din .. 2*din)
  const float* __restrict__ arow2 = arow + din;
  for (int k0 = 0; k0 < din; k0 += 4) {
    const float2 av = *(const float2*)(arow2 + k0);
    v2f a; a.x = av.x; a.y = av.y;
    const int kk = k0 + 2 * half;
    v2f b;
    b.x = Wr[(long long)kk * dout + n];
    b.y = Wr[(long long)(kk + 1) * dout + n];
    c = __builtin_amdgcn_wmma_f32_16x16x4_f32(false, a, false, b,
                                              (short)0, c, false, false);
  }

  const float bv = bias[n];
#pragma unroll
  for (int r = 0; r < 8; ++r) {
    const int row = rowBase + r + 8 * half;
    out[(long long)row * dout + n] = c[r] + bv;
  }
}

// --------------------------------------------- column sums / sum-of-squares
__global__ void sage_colstats_kernel(const float* __restrict__ h,
                                     float* __restrict__ stats,
                                     int N, int rowsPer) {
  const int col = threadIdx.x;                // blockDim.x == 256 == DHID
  int r0 = blockIdx.x * rowsPer;
  int r1 = r0 + rowsPer;
  if (r1 > N) r1 = N;
  float s = 0.0f, ss = 0.0f;
  for (int r = r0; r < r1; ++r) {
    const float v = h[(long long)r * DHID + col];
    s += v;
    ss += v * v;
  }
  atomicAdd(&stats[col], s);
  atomicAdd(&stats[DHID + col], ss);
}

// --------------------------------------------- BN (training mode) + ReLU
__global__ void sage_bnrelu_kernel(const float* __restrict__ hp,
                                   const float* __restrict__ stats,
                                   const float* __restrict__ g,
                                   const float* __restrict__ beta,
                                   float* __restrict__ out,
                                   long long total, float invN) {
  long long i = (long long)blockIdx.x * blockDim.x + threadIdx.x;
  long long stride = (long long)gridDim.x * blockDim.x;
  for (; i < total; i += stride) {
    const int col = (int)(i & (DHID - 1));
    const float mean = stats[col] * invN;
    const float var  = stats[DHID + col] * invN - mean * mean;
    const float sc = g[col] * rsqrtf(var + 1e-5f);
    const float sh = beta[col] - mean * sc;
    const float v = hp[i] * sc + sh;
    out[i] = fmaxf(v, 0.0f);
  }
}

// ---------------------------------------------------------------- launcher
extern "C" void kernel_launch(void* const* d_in, const int* in_sizes, int n_in,
                              void* d_out, int out_size, void* d_ws, size_t ws_size,
                              hipStream_t stream) {
  const float* x     = (const float*)d_in[0];
  const int*   esrc  = (const int*)d_in[1];
  const int*   edst  = (const int*)d_in[2];
  const float* Wl1   = (const float*)d_in[3];
  const float* bl1   = (const float*)d_in[4];
  const float* Wr1   = (const float*)d_in[5];
  const float* g1    = (const float*)d_in[6];
  const float* beta1 = (const float*)d_in[7];
  const float* Wl2   = (const float*)d_in[8];
  const float* bl2   = (const float*)d_in[9];
  const float* Wr2   = (const float*)d_in[10];
  const float* g2    = (const float*)d_in[11];
  const float* beta2 = (const float*)d_in[12];
  const float* Wl3   = (const float*)d_in[13];
  const float* bl3   = (const float*)d_in[14];
  const float* Wr3   = (const float*)d_in[15];

  const int N = in_sizes[0] / 128;   // 100000 (divisible by 16)
  const int E = in_sizes[1];         // 1600000
  const int DIN = 128;

  float* ws     = (float*)d_ws;
  float* invdeg = ws;                               // N
  float* stats  = ws + N;                           // 2*DHID
  float* agg    = stats + 2 * DHID;                 // N*DHID
  float* hpre   = agg  + (long long)N * DHID;       // N*DHID
  float* hbuf   = hpre + (long long)N * DHID;       // N*DHID
  float* outp   = (float*)d_out;                    // N*DHID

  const float invN = 1.0f / (float)N;
  const long long nd1 = (long long)N * DIN;
  const long long nd2 = (long long)N * DHID;

  dim3 gemmGrid((unsigned)(N / 16), DHID / 64);
  const int statBlocks = 400;
  const int rowsPer = (N + statBlocks - 1) / statBlocks;

  // degrees -> inverse degrees
  sage_zero_kernel<<<2048, 256, 0, stream>>>(invdeg, N);
  sage_degree_kernel<<<(E + 255) / 256, 256, 0, stream>>>(edst, invdeg, E);
  sage_invdeg_kernel<<<(N + 255) / 256, 256, 0, stream>>>(invdeg, N);

  // ---------------- layer 1 (x: N x 128 -> h: N x 256) ----------------
  sage_zero_kernel<<<4096, 256, 0, stream>>>(agg, nd1);
  {
    long long total = (long long)E << 5;  // chunks = 128/4 = 32
    sage_scatter_kernel<<<(unsigned)((total + 255) / 256), 256, 0, stream>>>(
        x, esrc, edst, agg, total, DIN, 5);
  }
  sage_gemm_wmma_kernel<<<gemmGrid, 128, 0, stream>>>(
      agg, x, invdeg, Wl1, Wr1, bl1, hpre, DIN, DHID, 8);
  sage_zero_kernel<<<2, 256, 0, stream>>>(stats, 2 * DHID);
  sage_colstats_kernel<<<statBlocks, DHID, 0, stream>>>(hpre, stats, N, rowsPer);
  sage_bnrelu_kernel<<<4096, 256, 0, stream>>>(hpre, stats, g1, beta1, hbuf, nd2, invN);

  // ---------------- layer 2 (h: N x 256 -> h: N x 256) ----------------
  sage_zero_kernel<<<4096, 256, 0, stream>>>(agg, nd2);
  {
    long long total = (long long)E << 6;  // chunks = 256/4 = 64
    sage_scatter_kernel<<<(unsigned)((total + 255) / 256), 256, 0, stream>>>(
        hbuf, esrc, edst, agg, total, DHID, 6);
  }
  sage_gemm_wmma_kernel<<<gemmGrid, 128, 0, stream>>>(
      agg, hbuf, invdeg, Wl2, Wr2, bl2, hpre, DHID, DHID, 9);
  sage_zero_kernel<<<2, 256, 0, stream>>>(stats, 2 * DHID);
  sage_colstats_kernel<<<statBlocks, DHID, 0, stream>>>(hpre, stats, N, rowsPer);
  sage_bnrelu_kernel<<<4096, 256, 0, stream>>>(hpre, stats, g2, beta2, hbuf, nd2, invN);

  // ---------------- layer 3 (h: N x 256 -> out: N x 256, no BN) -------
  sage_zero_kernel<<<4096, 256, 0, stream>>>(agg, nd2);
  {
    long long total = (long long)E << 6;
    sage_scatter_kernel<<<(unsigned)((total + 255) / 256), 256, 0, stream>>>(
        hbuf, esrc, edst, agg, total, DHID, 6);
  }
  sage_gemm_wmma_kernel<<<gemmGrid, 128, 0, stream>>>(
      agg, hbuf, invdeg, Wl3, Wr3, bl3, outp, DHID, DHID, 9);
}